// NeighborhoodAttention_6923487282074
// MI455X (gfx1250) — compile-verified
//
#include <hip/hip_runtime.h>
#include <stdint.h>

typedef __attribute__((ext_vector_type(2))) float v2f;
typedef __attribute__((ext_vector_type(8))) float v8f;

#define WMMA_F32(A, B, C) \
  __builtin_amdgcn_wmma_f32_16x16x4_f32(false, (A), false, (B), (short)0, (C), false, false)

static constexpr int EMB  = 256;   // embedding dim
static constexpr int KN   = 64;    // neighbors
static constexpr int BTOT = 16384; // batch

__device__ __forceinline__ v8f zero8() {
  v8f z = {0.f, 0.f, 0.f, 0.f, 0.f, 0.f, 0.f, 0.f};
  return z;
}

// Async DMA: 16 bytes global -> LDS, tracked by ASYNCcnt (no VGPR data movement).
// LDS dest must be 16B aligned. Low 32 bits of a flat shared pointer are the LDS addr.
__device__ __forceinline__ void async_ld_b128(void* lds_dst, const void* gsrc) {
  asm volatile("global_load_async_to_lds_b128 %0, %1, off"
               :: "v"((uint32_t)(uintptr_t)lds_dst), "v"(gsrc)
               : "memory");
}
__device__ __forceinline__ void wait_async0() {
  asm volatile("s_wait_asynccnt 0x0" ::: "memory");
}

// ---------------------------------------------------------------------------
// Kernel 1: v[b,:] = 0.125 * (center[b,:] @ Wq^T) @ Wk        [B,256]
// Block = 256 threads (8 waves). Each wave owns a 16-row M-tile -> 128 rows/block.
// ---------------------------------------------------------------------------
__global__ __launch_bounds__(256) void k1_qv(const float* __restrict__ center,
                                             const float* __restrict__ Wq,
                                             const float* __restrict__ Wk,
                                             float* __restrict__ vout) {
  extern __shared__ float lds1[];
  float* cen = lds1;              // [128][260]  center tile (stride 260 = 1040B, 16B aligned)
  float* qs  = lds1 + 128 * 260;  // [8][16][68] q tiles (D-layout -> A-layout reshuffle)

  const int t  = threadIdx.x;
  const int b0 = blockIdx.x * 128;

  // Stage 128 center rows via async global->LDS DMA (coalesced 16B per lane).
  for (int i = 0; i < 32; ++i) {
    int e = i * 1024 + t * 4;
    int r = e >> 8, c = e & 255;           // c % 4 == 0 -> 16B-aligned LDS dest
    async_ld_b128(cen + r * 260 + c, center + (size_t)(b0 + r) * EMB + c);
  }
  wait_async0();
  __syncthreads();

  const int w    = t >> 5;
  const int l    = t & 31;
  const int rowl = l & 15;         // A/B lane row (M=0..15 in both lane halves)
  const int koff = (l >> 4) * 2;   // K sub-offset per lane half
  const int rsel = (l >> 4) * 8;   // D/C row offset per lane half

  // ---- Q = center_tile @ Wq^T  (16 x 64, K=256) ----
  v8f accq[4];
#pragma unroll
  for (int nt = 0; nt < 4; ++nt) accq[nt] = zero8();

  for (int k0 = 0; k0 < EMB; k0 += 4) {
    v2f a;
    const float* ap = cen + (w * 16 + rowl) * 260 + k0 + koff;
    a.x = ap[0]; a.y = ap[1];
#pragma unroll
    for (int nt = 0; nt < 4; ++nt) {
      int n = nt * 16 + rowl;
      v2f bm;
      bm.x = Wq[(size_t)n * EMB + k0 + koff];
      bm.y = Wq[(size_t)n * EMB + k0 + koff + 1];
      accq[nt] = WMMA_F32(a, bm, accq[nt]);
    }
  }

  // Spill q (D-layout) to LDS so it can be re-read in A-layout.
#pragma unroll
  for (int nt = 0; nt < 4; ++nt)
#pragma unroll
    for (int j = 0; j < 8; ++j)
      qs[(w * 16 + j + rsel) * 68 + nt * 16 + rowl] = accq[nt][j];
  __syncthreads();

  // ---- V = Q @ Wk  (16 x 256, K=64), N split into halves of 128 ----
  for (int h = 0; h < 2; ++h) {
    v8f accv[8];
#pragma unroll
    for (int nt = 0; nt < 8; ++nt) accv[nt] = zero8();

    for (int k0 = 0; k0 < 64; k0 += 4) {
      v2f a;
      const float* ap = qs + (w * 16 + rowl) * 68 + k0 + koff;
      a.x = ap[0]; a.y = ap[1];
#pragma unroll
      for (int nt = 0; nt < 8; ++nt) {
        int n = h * 128 + nt * 16 + rowl;
        v2f bm;
        bm.x = Wk[(size_t)(k0 + koff) * EMB + n];      // Wk[a][d] row-major: B[k=a][n=d]
        bm.y = Wk[(size_t)(k0 + koff + 1) * EMB + n];
        accv[nt] = WMMA_F32(a, bm, accv[nt]);
      }
    }
#pragma unroll
    for (int nt = 0; nt < 8; ++nt)
#pragma unroll
      for (int j = 0; j < 8; ++j) {
        int gb  = b0 + w * 16 + j + rsel;
        int col = h * 128 + nt * 16 + rowl;
        vout[(size_t)gb * EMB + col] = accv[nt][j] * 0.125f;  // attn scale folded in
      }
  }
}

// ---------------------------------------------------------------------------
// Kernel 2: attention. One block per b. Single pass over neigh (1 GiB total),
// staged into LDS with async DMA (ASYNCcnt path).
//   logits[k] = neigh[b,k,:] . v[b,:]  -> softmax -> *ppi -> renorm -> context
// context written into d_out (k3 reads-then-overwrites elementwise).
// ---------------------------------------------------------------------------
__global__ __launch_bounds__(256) void k2_attn(const float* __restrict__ neigh,
                                               const float* __restrict__ v,
                                               const float* __restrict__ ppi,
                                               float* __restrict__ ctx) {
  extern __shared__ float lds2[];
  float* tile = lds2;              // [64][260]  neighbor tile (stride 260: 16B-aligned rows)
  float* vbuf = tile + 64 * 260;   // 256
  float* part = vbuf + 256;        // 256 partial dots
  float* sa   = part + 256;        // 64
  float* sb   = sa + 64;           // 64

  const int t = threadIdx.x;
  const int b = blockIdx.x;
  const float* nb = neigh + (size_t)b * (KN * EMB);

  // Stage 64x256 tile: async global->LDS B128 DMA, 16B per lane per issue.
#pragma unroll
  for (int i = 0; i < 16; ++i) {
    int e = i * 1024 + t * 4;
    int k = e >> 8, d = e & 255;           // d % 4 == 0 -> aligned
    async_ld_b128(tile + k * 260 + d, nb + e);
  }
  vbuf[t] = v[(size_t)b * EMB + t];
  wait_async0();
  __syncthreads();

  // Partial logits: thread (c,k) dots a 64-wide chunk of row k with v.
  {
    int c = t >> 6, k = t & 63;
    const float* tr = tile + k * 260 + c * 64;
    const float* vr = vbuf + c * 64;  // same addr across lanes -> LDS broadcast
    float s = 0.f;
#pragma unroll 8
    for (int j = 0; j < 64; ++j) s += tr[j] * vr[j];
    part[t] = s;
  }
  __syncthreads();
  if (t < 64) sa[t] = part[t] + part[64 + t] + part[128 + t] + part[192 + t];
  __syncthreads();

  // Softmax, ppi reweight, renormalize (exact reference ordering).
  if (t < 64) {
    float m = -3.4e38f;
    for (int j = 0; j < 64; ++j) m = fmaxf(m, sa[j]);
    sb[t] = __expf(sa[t] - m);
  }
  __syncthreads();
  if (t < 64) {
    float s1 = 0.f;
    for (int j = 0; j < 64; ++j) s1 += sb[j];
    sa[t] = (sb[t] / s1) * ppi[(size_t)b * 64 + t];
  }
  __syncthreads();
  if (t < 64) {
    float s2 = 0.f;
    for (int j = 0; j < 64; ++j) s2 += sa[j];
    sb[t] = sa[t] / (s2 + 1e-8f);
  }
  __syncthreads();

  // context[d] = sum_k attn[k] * tile[k][d]   (bank = (4k+d)%64: conflict-free)
  {
    float acc = 0.f;
#pragma unroll 8
    for (int k = 0; k < 64; ++k) acc += sb[k] * tile[k * 260 + t];
    ctx[(size_t)b * EMB + t] = acc;
  }
}

// ---------------------------------------------------------------------------
// Kernel 3: gate = sigmoid([center|ctx] @ Wgate^T + bgate); out = blend.
// Block = 16-row M-tile; 8 waves x 2 N-tiles each cover N=256; K=512 via WMMA.
// ctx aliases out: each element is read (staging + epilogue) before its single write.
// ---------------------------------------------------------------------------
__global__ __launch_bounds__(256) void k3_gate(const float* __restrict__ center,
                                               const float* ctx,
                                               const float* __restrict__ Wgate,
                                               const float* __restrict__ bgate,
                                               float* out) {
  __shared__ float Xs[16 * 516];  // [16][516] (2064B rows, 16B aligned; conflict-free A reads)
  const int t  = threadIdx.x;
  const int b0 = blockIdx.x * 16;

  // Stage X = [center | ctx] tile via async DMA, 4 floats per issue.
  for (int i = 0; i < 8; ++i) {
    int e = (i * 256 + t) * 4;             // 0..8188, step 4
    int r = e >> 9, c = e & 511;           // c % 4 == 0: never straddles the 256 split
    const float* src = (c < 256) ? center + (size_t)(b0 + r) * 256 + c
                                 : ctx + (size_t)(b0 + r) * 256 + (c - 256);
    async_ld_b128(Xs + r * 516 + c, src);
  }
  wait_async0();
  __syncthreads();

  const int w    = t >> 5;
  const int l    = t & 31;
  const int rowl = l & 15;
  const int koff = (l >> 4) * 2;
  const int rsel = (l >> 4) * 8;

  v8f acc0 = zero8(), acc1 = zero8();
  for (int k0 = 0; k0 < 512; k0 += 4) {
    v2f a;
    const float* ap = Xs + rowl * 516 + k0 + koff;
    a.x = ap[0]; a.y = ap[1];
    {
      int n = (w * 2 + 0) * 16 + rowl;
      v2f bm;
      bm.x = Wgate[(size_t)n * 512 + k0 + koff];
      bm.y = Wgate[(size_t)n * 512 + k0 + koff + 1];
      acc0 = WMMA_F32(a, bm, acc0);
    }
    {
      int n = (w * 2 + 1) * 16 + rowl;
      v2f bm;
      bm.x = Wgate[(size_t)n * 512 + k0 + koff];
      bm.y = Wgate[(size_t)n * 512 + k0 + koff + 1];
      acc1 = WMMA_F32(a, bm, acc1);
    }
  }

#pragma unroll
  for (int q = 0; q < 2; ++q) {
    v8f acc = q ? acc1 : acc0;
    int n = (w * 2 + q) * 16 + rowl;
    float bg = bgate[n];
#pragma unroll
    for (int j = 0; j < 8; ++j) {
      int bb = b0 + j + rsel;
      size_t idx = (size_t)bb * 256 + n;
      float g  = 1.f / (1.f + __expf(-(acc[j] + bg)));
      float ce = center[idx];
      float cx = ctx[idx];          // read before the aliased write below
      out[idx] = g * ce + (1.f - g) * cx;
    }
  }
}

// ---------------------------------------------------------------------------
extern "C" void kernel_launch(void* const* d_in, const int* in_sizes, int n_in,
                              void* d_out, int out_size, void* d_ws, size_t ws_size,
                              hipStream_t stream) {
  const float* center = (const float*)d_in[0];
  const float* neigh  = (const float*)d_in[1];
  const float* ppi    = (const float*)d_in[2];
  const float* Wq     = (const float*)d_in[3];
  const float* Wk     = (const float*)d_in[4];
  const float* Wgate  = (const float*)d_in[5];
  const float* bgate  = (const float*)d_in[6];

  float* out = (float*)d_out;
  float* vws = (float*)d_ws;  // v[B,256] = 16 MB scratch
  float* ctx = out;           // context staged in d_out, overwritten by k3

  const size_t lds1 = (size_t)(128 * 260 + 8 * 16 * 68) * sizeof(float);
  const size_t lds2 = (size_t)(64 * 260 + 256 + 256 + 64 + 64) * sizeof(float);

  hipLaunchKernelGGL(k1_qv,   dim3(BTOT / 128), dim3(256), lds1, stream,
                     center, Wq, Wk, vws);
  hipLaunchKernelGGL(k2_attn, dim3(BTOT),       dim3(256), lds2, stream,
                     neigh, vws, ppi, ctx);
  hipLaunchKernelGGL(k3_gate, dim3(BTOT / 16),  dim3(256), 0,    stream,
                     center, ctx, Wgate, bgate, out);
}